// ConvMod_47614007443904
// MI455X (gfx1250) — compile-verified
//
#include <hip/hip_runtime.h>
#include <stdint.h>

#define IMG_H 4096
#define IMG_W 4096

typedef __attribute__((ext_vector_type(2))) float v2f;
typedef __attribute__((ext_vector_type(8))) float v8f;

// Tile geometry: flat 256-thread block = 8 wave32s; tile = 128 cols x 16 rows.
constexpr int TILE_W = 128;                 // output cols per block
constexpr int TILE_H = 16;                  // output rows per block
constexpr int LDS_ROWS = TILE_H + 2;        // 18: X rows gp0-1 .. gp0+16
constexpr int LDS_W = TILE_W + 8;           // 136 floats: 4-col halo each side (16B aligned)
constexpr int CHUNKS_PER_ROW = LDS_W / 4;   // 34 float4 chunks per row
constexpr int TOTAL_CHUNKS = LDS_ROWS * CHUNKS_PER_ROW;  // 612
constexpr int GRID_X = IMG_W / TILE_W;      // 32
constexpr int GRID_Y = IMG_H / TILE_H;      // 256

// Select E[u+1][v] for u in {-1,0,1}, else 0 (band matrix entry).
__device__ __forceinline__ float band_pick(int u, float e0, float e1, float e2) {
  float r = 0.0f;
  r = (u == -1) ? e0 : r;
  r = (u == 0) ? e1 : r;
  r = (u == 1) ? e2 : r;
  return r;
}

__global__ __launch_bounds__(256, 4) void fused_conv2x2_tconv2x2(
    const float* __restrict__ X, const float* __restrict__ Wk,
    float* __restrict__ Y) {
  __shared__ __align__(16) float tile[LDS_ROWS][LDS_W];

  const int bx = blockIdx.x;
  const int by = blockIdx.y;
  const int tid = threadIdx.x;           // 0..255

  const int gp0 = by * TILE_H;           // first output row of this block
  const int q0 = bx * TILE_W;            // first output col of this block
  const int qbase = q0 - 4;              // global col corresponding to LDS col 0

  // Low 32 bits of a generic pointer to LDS == wave-relative LDS byte address.
  const uint32_t lds_base = (uint32_t)(uintptr_t)(&tile[0][0]);

  // ---- Stage X tile (with halo) into LDS via async global->LDS 16B copies ----
  // Edge blocks clamp the source address; clamped data is finite and is only
  // ever multiplied by a 0 boundary mask / 0 band entry below.
  for (int idx = tid; idx < TOTAL_CHUNKS; idx += 256) {
    const int r = idx / CHUNKS_PER_ROW;
    const int c4 = idx - r * CHUNKS_PER_ROW;
    int grow = gp0 - 1 + r;
    grow = grow < 0 ? 0 : (grow > IMG_H - 1 ? IMG_H - 1 : grow);
    int gcol = qbase + c4 * 4;
    gcol = gcol < 0 ? 0 : (gcol > IMG_W - 4 ? IMG_W - 4 : gcol);
    const uint32_t lds_addr =
        lds_base + (uint32_t)((r * LDS_W + c4 * 4) * (int)sizeof(float));
    const uint64_t gaddr =
        (uint64_t)(uintptr_t)(X + (size_t)grow * IMG_W + gcol);
    asm volatile("global_load_async_to_lds_b128 %0, %1, off"
                 :: "v"(lds_addr), "v"(gaddr)
                 : "memory");
  }

#if __has_builtin(__builtin_amdgcn_s_wait_asynccnt)
  __builtin_amdgcn_s_wait_asynccnt(0);
#else
  asm volatile("s_wait_asynccnt 0x0" ::: "memory");
#endif
  __syncthreads();

  const float w00 = Wk[0], w01 = Wk[1], w10 = Wk[2], w11 = Wk[3];

  const bool interior = (by > 0) && (by < GRID_Y - 1) && (bx > 0) && (bx < GRID_X - 1);

  if (interior) {
    // ================= WMMA path (all boundary masks == 1) =================
    // Fused op == 3x3 stencil with E = autocorr(W):
    //   y2[p,q] = sum_{u,v} E[u+1][v+1] * X[p+u][q+v]
    // Per 16x16 tile: Y = sum_{v=0..2} BandA_v(16x18) * Xshift_v(18x16),
    // each banded matmul done as 5 chunks of V_WMMA_F32_16X16X4_F32.
    const float E00 = w11 * w00;
    const float E01 = w10 * w00 + w11 * w01;
    const float E02 = w10 * w01;
    const float E10 = w01 * w00 + w11 * w10;
    const float E11 = w00 * w00 + w01 * w01 + w10 * w10 + w11 * w11;
    const float E12 = w00 * w01 + w10 * w11;
    const float E20 = w01 * w10;
    const float E21 = w00 * w10 + w01 * w11;
    const float E22 = w00 * w11;

    const int lane = tid & 31;
    const int wid = tid >> 5;            // 0..7: which 16-col tile
    const int m = lane & 15;             // A row / D col index
    const int kh = lane >> 4;            // lane half: K offset 0 or 2
    const int k0 = kh << 1;
    const int ncol = lane & 15;          // B/D column within tile
    const int ltile0 = wid * 16;         // tile-local col offset in block

    v8f acc = {};
#pragma unroll
    for (int v = 0; v < 3; ++v) {
      const float e0 = (v == 0) ? E00 : ((v == 1) ? E01 : E02);
      const float e1 = (v == 0) ? E10 : ((v == 1) ? E11 : E12);
      const float e2 = (v == 0) ? E20 : ((v == 1) ? E21 : E22);
      // LDS col of X[.][q0 + ltile0 + ncol + v - 1]
      const int lcol = ltile0 + 3 + v + ncol;
#pragma unroll
      for (int c = 0; c < 5; ++c) {
        const int r0 = 4 * c + k0;       // band row handled by this lane, VGPR 0
        const int r1 = r0 + 1;           // band row handled by this lane, VGPR 1
        // A[m][k] = E[u+1][v] with u = r - 1 - m (0 outside the band / r>=18)
        v2f a, b;
        a.x = band_pick(r0 - 1 - m, e0, e1, e2);
        a.y = band_pick(r1 - 1 - m, e0, e1, e2);
        // B[k][n] = X[gp0-1+r][...]; rows >=18 have A==0, clamp read in-range.
        const int rr0 = r0 > LDS_ROWS - 1 ? LDS_ROWS - 1 : r0;
        const int rr1 = r1 > LDS_ROWS - 1 ? LDS_ROWS - 1 : r1;
        b.x = tile[rr0][lcol];
        b.y = tile[rr1][lcol];
        acc = __builtin_amdgcn_wmma_f32_16x16x4_f32(
            false, a, false, b, (short)0, acc, false, false);
      }
    }

    // D layout: VGPR j -> row gp0 + j + 8*kh, col q0 + ltile0 + ncol.
    float* yrow = Y + (size_t)(gp0 + kh * 8) * IMG_W + (q0 + ltile0 + ncol);
#pragma unroll
    for (int j = 0; j < 8; ++j) {
      __builtin_nontemporal_store(acc[j], yrow + (size_t)j * IMG_W);
    }
  } else {
    // ============ VALU path: exact boundary-clipped semantics ============
    const int tx = tid & 127;            // one output column each
    const int ty = tid >> 7;             // 8 output rows each

    const int q = q0 + tx;
    const float mb0 = (q <= IMG_W - 2) ? 1.0f : 0.0f;  // b=0 tap valid
    const float mb1 = (q >= 1) ? 1.0f : 0.0f;          // b=1 tap valid

    const int pbase = gp0 + ty * 8;

    // xr[r][v] = X[pbase-1+r][q-1+v]; lanes read consecutive LDS cols.
    float xr[10][3];
#pragma unroll
    for (int r = 0; r < 10; ++r) {
#pragma unroll
      for (int v = 0; v < 3; ++v) {
        xr[r][v] = tile[ty * 8 + r][tx + 3 + v];
      }
    }

#pragma unroll
    for (int k = 0; k < 8; ++k) {
      const int p = pbase + k;
      const float ma0 = (p <= IMG_H - 2) ? 1.0f : 0.0f;
      const float ma1 = (p >= 1) ? 1.0f : 0.0f;
      const float wm00 = ma0 * mb0 * w00;
      const float wm01 = ma0 * mb1 * w01;
      const float wm10 = ma1 * mb0 * w10;
      const float wm11 = ma1 * mb1 * w11;

      const float x00 = xr[k][0], x01 = xr[k][1], x02 = xr[k][2];
      const float x10 = xr[k + 1][0], x11 = xr[k + 1][1], x12 = xr[k + 1][2];
      const float x20 = xr[k + 2][0], x21 = xr[k + 2][1], x22 = xr[k + 2][2];

      const float y1_00 = w00 * x11 + w01 * x12 + w10 * x21 + w11 * x22;
      const float y1_01 = w00 * x10 + w01 * x11 + w10 * x20 + w11 * x21;
      const float y1_10 = w00 * x01 + w01 * x02 + w10 * x11 + w11 * x12;
      const float y1_11 = w00 * x00 + w01 * x01 + w10 * x10 + w11 * x11;

      const float out =
          wm00 * y1_00 + wm01 * y1_01 + wm10 * y1_10 + wm11 * y1_11;
      __builtin_nontemporal_store(out, &Y[(size_t)p * IMG_W + q]);
    }
  }
}

extern "C" void kernel_launch(void* const* d_in, const int* in_sizes, int n_in,
                              void* d_out, int out_size, void* d_ws,
                              size_t ws_size, hipStream_t stream) {
  (void)in_sizes; (void)n_in; (void)d_ws; (void)ws_size; (void)out_size;
  const float* X = (const float*)d_in[0];   // [4096,4096] fp32
  const float* Wk = (const float*)d_in[1];  // [2,2] fp32, row-major
  float* Y = (float*)d_out;                 // [4096,4096] fp32

  dim3 grid(GRID_X, GRID_Y);                // (32, 256)
  dim3 block(256);                          // 8 wave32s; one 16x16 WMMA tile/wave
  fused_conv2x2_tconv2x2<<<grid, block, 0, stream>>>(X, Wk, Y);
}